// GNNEncoder_5454608466351
// MI455X (gfx1250) — compile-verified
//
#include <hip/hip_runtime.h>

typedef float v2f __attribute__((ext_vector_type(2)));
typedef float v8f __attribute__((ext_vector_type(8)));

#define D 128
#define MS (D * D)  // elements per weight matrix
#define EPS 1e-5f
#define SLOPE 0.01f
#define APAD 132  // padded LDS row stride (floats): conflict-free b64 reads, 16B-aligned rows

// ---------------------------------------------------------------- utilities

__global__ void zero_f32(float* __restrict__ p, long n) {
  long i = (long)blockIdx.x * blockDim.x + threadIdx.x;
  long stride = (long)gridDim.x * blockDim.x;
  for (; i < n; i += stride) p[i] = 0.0f;
}

// one edge per 128-thread block: coalesced gather of source row, fp32 atomic
// scatter into destination accumulator (L2-resident tables -> L2 atomics).
__global__ void scatter_add(const float* __restrict__ xsrc,
                            const int* __restrict__ src,
                            const int* __restrict__ dst,
                            float* __restrict__ acc,
                            float* __restrict__ cnt,
                            int E) {
  int e = blockIdx.x;
  if (e >= E) return;
  int s = src[e];
  int d = dst[e];
  int c = threadIdx.x;  // 0..127
  float v = xsrc[(long)s * D + c];
  atomicAdd(&acc[(long)d * D + c], v);
  if (c == 0) atomicAdd(&cnt[d], 1.0f);
}

__global__ void seg_div(float* __restrict__ acc, const float* __restrict__ cnt,
                        int n) {
  long i = (long)blockIdx.x * blockDim.x + threadIdx.x;
  if (i >= (long)n * D) return;
  float c = cnt[i >> 7];
  acc[i] = acc[i] / fmaxf(c, 1.0f);
}

// repack W[k][col] (row-major 128x128) into K-pair-interleaved layout:
// Wp[(k>>1)*256 + col*2 + (k&1)] so a WMMA B fragment (W[ka][c], W[ka+1][c])
// is one contiguous 8-byte load, coalesced across lanes.
__global__ void repack_w(const float* __restrict__ W, float* __restrict__ Wp) {
  int i = blockIdx.x * blockDim.x + threadIdx.x;  // 0..16383
  int k = i >> 7;
  int col = i & 127;
  Wp[(k >> 1) * 256 + col * 2 + (k & 1)] = W[i];
}

// ---------------------------------------------------------------- WMMA GEMM

__device__ __forceinline__ v8f wmma4(v2f a, v2f b, v8f c) {
  // D = A(16x4 f32) * B(4x16 f32) + C(16x16 f32)
  return __builtin_amdgcn_wmma_f32_16x16x4_f32(
      /*neg_a=*/false, a, /*neg_b=*/false, b,
      /*c_mod=*/(short)0, c, /*reuse_a=*/false, /*reuse_b=*/false);
}

// PW = 6 packed weight matrices for this layer, contiguous:
//   [0]=Wn_cites [1]=Wr_cites [2]=Wn_writes [3]=Wr_writes [4]=Wn_rev [5]=Wr_rev
// out[M x 128] = Ac@Wn_c + Aw@Wn_w + Xp@Wr_c + Xp@Wr_w + (bc + bw)
// 256 threads = 8 waves; wave w owns cols [16w,16w+16); block owns 16 rows.
// A tiles staged once per block into padded LDS (conflict-free ds_load_b64);
// B via one running pointer + compile-time immediate offsets.
__global__ void gemm_paper(const float* __restrict__ Ac,
                           const float* __restrict__ Aw,
                           const float* __restrict__ Xp,
                           const float* __restrict__ PW,
                           const float* __restrict__ bc,
                           const float* __restrict__ bw,
                           float* __restrict__ out, int M) {
  __shared__ float sA[3 * 16 * APAD];
  int row0 = blockIdx.x * 16;
  int t = threadIdx.x;

  // cooperative stage: three 16x128 tiles, coalesced float4 row loads
  {
    const float* srcs[3] = {Ac, Aw, Xp};
#pragma unroll
    for (int m = 0; m < 3; ++m) {
      const float* src = srcs[m] + (long)row0 * D;
      float* dstm = sA + m * (16 * APAD);
      for (int idx = t; idx < 512; idx += 256) {
        int row = idx >> 5;
        int c4 = (idx & 31) * 4;
        *(float4*)(dstm + row * APAD + c4) = *(const float4*)(src + row * D + c4);
      }
    }
  }
  __syncthreads();

  int wave = t >> 5;
  int lane = t & 31;
  int col0 = wave * 16;
  int r  = lane & 15;   // A row within tile
  int kh = lane >> 4;   // K-half: 0 -> K{0,1}, 1 -> K{2,3}
  int c  = lane & 15;   // B/C col within tile

  // single LDS base; three tiles reached via immediate offsets
  const float* pA = sA + r * APAD + 2 * kh;
  // single global base for all four B matrices; immediate offsets 0/MS/2MS/3MS
  const float* pB = PW + (col0 + c) * 2 + kh * 256;

  v8f acc0 = {}; v8f acc1 = {}; v8f acc2 = {}; v8f acc3 = {};
#pragma unroll 4
  for (int kk = 0; kk < 32; ++kk) {
    v2f a_c = *(const v2f*)(pA);
    v2f a_w = *(const v2f*)(pA + 16 * APAD);
    v2f a_x = *(const v2f*)(pA + 32 * APAD);
    v2f b0 = *(const v2f*)(pB);            // Wn_cites
    v2f b1 = *(const v2f*)(pB + 2 * MS);   // Wn_writes
    v2f b2 = *(const v2f*)(pB + 1 * MS);   // Wr_cites
    v2f b3 = *(const v2f*)(pB + 3 * MS);   // Wr_writes
    acc0 = wmma4(a_c, b0, acc0);   // 4 independent accumulation chains
    acc1 = wmma4(a_w, b1, acc1);
    acc2 = wmma4(a_x, b2, acc2);
    acc3 = wmma4(a_x, b3, acc3);
    pA += 4;    // next K-group in LDS
    pB += 512;  // next K-pair block in packed weights
  }
  float bias = bc[col0 + c] + bw[col0 + c];
#pragma unroll
  for (int g = 0; g < 8; ++g) {
    int row = row0 + g + 8 * kh;
    out[(long)row * D + col0 + c] = acc0[g] + acc1[g] + acc2[g] + acc3[g] + bias;
  }
}

// out[M x 128] = Ar@Wn_rev + Xa@Wr_rev + b   (PW as above, matrices 4 and 5)
__global__ void gemm_author(const float* __restrict__ Ar,
                            const float* __restrict__ Xa,
                            const float* __restrict__ PW,
                            const float* __restrict__ bb,
                            float* __restrict__ out, int M) {
  __shared__ float sA[2 * 16 * APAD];
  int row0 = blockIdx.x * 16;
  int t = threadIdx.x;
  {
    const float* srcs[2] = {Ar, Xa};
#pragma unroll
    for (int m = 0; m < 2; ++m) {
      const float* src = srcs[m] + (long)row0 * D;
      float* dstm = sA + m * (16 * APAD);
      for (int idx = t; idx < 512; idx += 256) {
        int row = idx >> 5;
        int c4 = (idx & 31) * 4;
        *(float4*)(dstm + row * APAD + c4) = *(const float4*)(src + row * D + c4);
      }
    }
  }
  __syncthreads();

  int wave = t >> 5;
  int lane = t & 31;
  int col0 = wave * 16;
  int r  = lane & 15;
  int kh = lane >> 4;
  int c  = lane & 15;

  const float* pA = sA + r * APAD + 2 * kh;
  const float* pB = PW + 4 * MS + (col0 + c) * 2 + kh * 256;

  v8f acc0 = {}; v8f acc1 = {};
#pragma unroll 4
  for (int kk = 0; kk < 32; ++kk) {
    v2f a_n = *(const v2f*)(pA);
    v2f a_x = *(const v2f*)(pA + 16 * APAD);
    v2f b0 = *(const v2f*)(pB);        // Wn_rev
    v2f b1 = *(const v2f*)(pB + MS);   // Wr_rev
    acc0 = wmma4(a_n, b0, acc0);
    acc1 = wmma4(a_x, b1, acc1);
    pA += 4;
    pB += 512;
  }
  float bias = bb[col0 + c];
#pragma unroll
  for (int g = 0; g < 8; ++g) {
    int row = row0 + g + 8 * kh;
    out[(long)row * D + col0 + c] = acc0[g] + acc1[g] + bias;
  }
}

// ---------------------------------------------------------------- BatchNorm

// stats[0..127] += column sums, stats[128..255] += column sums of squares
__global__ void bn_stats(const float* __restrict__ x, int n,
                         float* __restrict__ stats) {
  __shared__ float s_sum[256];
  __shared__ float s_sq[256];
  const int ROWS = 256;
  int t = threadIdx.x;
  int col = t & 127;
  int half = t >> 7;  // 0 or 1
  long row0 = (long)blockIdx.x * ROWS;
  long rend = row0 + ROWS;
  if (rend > n) rend = n;
  float sum = 0.0f, sq = 0.0f;
  for (long rr = row0 + half; rr < rend; rr += 2) {
    float v = x[rr * D + col];
    sum += v;
    sq += v * v;
  }
  s_sum[t] = sum;
  s_sq[t] = sq;
  __syncthreads();
  if (half == 0) {
    atomicAdd(&stats[col], s_sum[t] + s_sum[t + 128]);
    atomicAdd(&stats[128 + col], s_sq[t] + s_sq[t + 128]);
  }
}

// in-place: y = lrelu((x - m) * rsqrt(v + eps) * g + b)
__global__ void bn_lrelu(float* __restrict__ x, const float* __restrict__ stats,
                         const float* __restrict__ g, const float* __restrict__ b,
                         int n) {
  long i = (long)blockIdx.x * blockDim.x + threadIdx.x;
  if (i >= (long)n * D) return;
  int col = (int)(i & 127);
  float inv_n = 1.0f / (float)n;
  float m = stats[col] * inv_n;
  float var = stats[128 + col] * inv_n - m * m;  // biased variance
  float y = (x[i] - m) * rsqrtf(var + EPS) * g[col] + b[col];
  x[i] = (y >= 0.0f) ? y : SLOPE * y;
}

// ---------------------------------------------------------------- driver

static inline int cdiv(long a, long b) { return (int)((a + b - 1) / b); }

extern "C" void kernel_launch(void* const* d_in, const int* in_sizes, int n_in,
                              void* d_out, int out_size, void* d_ws,
                              size_t ws_size, hipStream_t stream) {
  const float* x_paper  = (const float*)d_in[0];
  const float* x_author = (const float*)d_in[1];
  const int*   ei_c     = (const int*)d_in[2];
  const int*   ei_w     = (const int*)d_in[3];
  const int*   ei_r     = (const int*)d_in[4];
  const int Ec = in_sizes[2] / 2;
  const int Ew = in_sizes[3] / 2;
  const int Er = in_sizes[4] / 2;
  const int NP = in_sizes[0] / D;   // 100000 (multiple of 16)
  const int NA = in_sizes[1] / D;   // 50000  (multiple of 16)

  // params flattened in setup_inputs() insertion order:
  // layers[l] -> {cites, writes, rev}; each -> (Wn, b, Wr); then bn paper(g,b), author(g,b)
  auto P = [&](int l, int rel, int which) -> const float* {
    return (const float*)d_in[5 + l * 9 + rel * 3 + which];
  };
  const float* bn_pg = (const float*)d_in[23];
  const float* bn_pb = (const float*)d_in[24];
  const float* bn_ag = (const float*)d_in[25];
  const float* bn_ab = (const float*)d_in[26];

  // workspace layout (floats)
  float* ws = (float*)d_ws;
  float* mean_c = ws;                              // NP*D
  float* mean_w = mean_c + (size_t)NP * D;         // NP*D
  float* mean_r = mean_w + (size_t)NP * D;         // NA*D
  float* cnt_c  = mean_r + (size_t)NA * D;         // NP
  float* cnt_w  = cnt_c + NP;                      // NP
  float* cnt_r  = cnt_w + NP;                      // NA
  float* xp1    = cnt_r + NA;                      // NP*D  (layer-0 hidden, in-place BN)
  float* xa1    = xp1 + (size_t)NP * D;            // NA*D
  float* stats_p = xa1 + (size_t)NA * D;           // 256
  float* stats_a = stats_p + 256;                  // 256
  float* wpk     = stats_a + 256;                  // 12 * MS packed weights

  // packed layout per layer l (contiguous, 6 matrices):
  //   [0]=Wn_c [1]=Wr_c [2]=Wn_w [3]=Wr_w [4]=Wn_r [5]=Wr_r
  auto PKL = [&](int l) -> float* { return wpk + (size_t)l * 6 * MS; };

  const long mean_span = 2L * NP * D + (long)NA * D + 2L * NP + NA;

  float* out_p = (float*)d_out;
  float* out_a = out_p + (size_t)NP * D;

  // repack all 12 weight matrices once (tiny, L2-resident)
  for (int l = 0; l < 2; ++l)
    for (int rel = 0; rel < 3; ++rel) {
      repack_w<<<MS / 256, 256, 0, stream>>>(P(l, rel, 0), PKL(l) + (size_t)(rel * 2 + 0) * MS);
      repack_w<<<MS / 256, 256, 0, stream>>>(P(l, rel, 2), PKL(l) + (size_t)(rel * 2 + 1) * MS);
    }

  for (int l = 0; l < 2; ++l) {
    const float* XP = (l == 0) ? x_paper : xp1;
    const float* XA = (l == 0) ? x_author : xa1;
    float* hp = (l == 0) ? xp1 : out_p;
    float* ha = (l == 0) ? xa1 : out_a;

    // 1) zero segment-mean accumulators + counts (contiguous region)
    zero_f32<<<4096, 256, 0, stream>>>(mean_c, mean_span);

    // 2) segment sums (edge-parallel fp32 atomics) + counts
    scatter_add<<<Ec, D, 0, stream>>>(XP, ei_c, ei_c + Ec, mean_c, cnt_c, Ec);
    scatter_add<<<Ew, D, 0, stream>>>(XA, ei_w, ei_w + Ew, mean_w, cnt_w, Ew);
    scatter_add<<<Er, D, 0, stream>>>(XP, ei_r, ei_r + Er, mean_r, cnt_r, Er);

    // 3) sums -> means
    seg_div<<<cdiv((long)NP * D, 256), 256, 0, stream>>>(mean_c, cnt_c, NP);
    seg_div<<<cdiv((long)NP * D, 256), 256, 0, stream>>>(mean_w, cnt_w, NP);
    seg_div<<<cdiv((long)NA * D, 256), 256, 0, stream>>>(mean_r, cnt_r, NA);

    // 4) fused hetero-SAGE GEMMs via v_wmma_f32_16x16x4_f32
    gemm_paper<<<NP / 16, 256, 0, stream>>>(
        mean_c, mean_w, XP, PKL(l),
        P(l, 0, 1), P(l, 1, 1),     // b cites, b writes
        hp, NP);
    gemm_author<<<NA / 16, 256, 0, stream>>>(
        mean_r, XA, PKL(l), P(l, 2, 1), ha, NA);

    // 5) BN + LeakyReLU between layers (in place)
    if (l == 0) {
      zero_f32<<<1, 256, 0, stream>>>(stats_p, 512);  // stats_p + stats_a
      bn_stats<<<cdiv(NP, 256), 256, 0, stream>>>(xp1, NP, stats_p);
      bn_stats<<<cdiv(NA, 256), 256, 0, stream>>>(xa1, NA, stats_a);
      bn_lrelu<<<cdiv((long)NP * D, 256), 256, 0, stream>>>(xp1, stats_p, bn_pg, bn_pb, NP);
      bn_lrelu<<<cdiv((long)NA * D, 256), 256, 0, stream>>>(xa1, stats_a, bn_ag, bn_ab, NA);
    }
  }
}